// GAN_51015621542484
// MI455X (gfx1250) — compile-verified
//
#include <hip/hip_runtime.h>

typedef __attribute__((ext_vector_type(2))) float v2f;
typedef __attribute__((ext_vector_type(8))) float v8f;

#define F_OUT 64
#define NEG_SLOPE 0.2f

// ---------------- utility: order-preserving float<->uint for atomicMax ----------
__device__ __forceinline__ unsigned enc_f(float x) {
    unsigned b = __float_as_uint(x);
    return (b & 0x80000000u) ? ~b : (b | 0x80000000u);
}
__device__ __forceinline__ float dec_f(unsigned e) {
    unsigned b = (e & 0x80000000u) ? (e & 0x7FFFFFFFu) : ~e;
    return __uint_as_float(b);
}

// ---------------- zero fill ------------------------------------------------------
__global__ void k_zero(float* p, long long n) {
    long long i = (long long)blockIdx.x * blockDim.x + threadIdx.x;
    if (i < n) p[i] = 0.0f;
}

// ---------------- fp32 WMMA GEMM: C[M,64] = A[M,K] * B[K,64] ---------------------
// One wave32 per 16x16 tile. blockDim = (32,4): 4 waves cover the 64-wide N dim.
// A-frag layout (16x4 f32): lanes 0-15 -> M=lm, K=k0..k0+1; lanes 16-31 -> K=k0+2..k0+3
// B-frag layout (4x16 f32): VGPR0 -> K=k0 (lanes 0-15) / K=k0+2 (lanes 16-31); VGPR1 -> +1
// C layout (16x16 f32, 8 VGPRs): VGPR r -> row r (lanes 0-15) / row r+8 (lanes 16-31)
__global__ void k_gemm_wmma(const float* __restrict__ A, const float* __restrict__ B,
                            float* __restrict__ C, int M, int K) {
    const int lane = threadIdx.x;        // 0..31
    const int half = lane >> 4;          // 0 or 1
    const int lm   = lane & 15;
    const int mtile = blockIdx.x;
    const int col   = threadIdx.y * 16 + lm;   // 0..63

    int row = mtile * 16 + lm;
    if (row >= M) row = M - 1;           // clamp loads (M is 16-aligned here anyway)

    v8f acc = {};
    for (int k0 = 0; k0 < K; k0 += 4) {
        const int ka = k0 + half * 2;
        v2f a, b;
        a.x = A[(long long)row * K + ka];
        a.y = A[(long long)row * K + ka + 1];
        b.x = B[(ka    ) * F_OUT + col];
        b.y = B[(ka + 1) * F_OUT + col];
        // 8 args: (neg_a, A, neg_b, B, c_mod, C, reuse_a, reuse_b)
        acc = __builtin_amdgcn_wmma_f32_16x16x4_f32(false, a, false, b,
                                                    (short)0, acc, false, false);
    }

    const int rbase = mtile * 16 + half * 8;
#pragma unroll
    for (int r = 0; r < 8; ++r) {
        const int orow = rbase + r;
        if (orow < M) C[(long long)orow * F_OUT + col] = acc[r];
    }
}

// ---------------- per-node attention logits: h . a_src, h . a_dst ---------------
__global__ void k_alpha(const float* __restrict__ H,
                        const float* __restrict__ a_s, const float* __restrict__ a_d,
                        float* __restrict__ outs, float* __restrict__ outd, int N) {
    int i = blockIdx.x * blockDim.x + threadIdx.x;
    if (i >= N) return;
    const float* h = H + (long long)i * F_OUT;
    float ss = 0.0f, sd = 0.0f;
#pragma unroll
    for (int f = 0; f < F_OUT; ++f) {
        float v = h[f];
        ss += v * a_s[f];
        sd += v * a_d[f];
    }
    outs[i] = ss;
    outd[i] = sd;
}

// ---------------- edge pass 1: e = leakyrelu(as[src]+ad[dst]); segment max ------
__global__ void k_edge_max(const int* __restrict__ src, const int* __restrict__ dst,
                           int E0, int ET,
                           const float* __restrict__ asr, const float* __restrict__ adt,
                           float* __restrict__ ebuf, unsigned* __restrict__ menc) {
    int e = blockIdx.x * blockDim.x + threadIdx.x;
    if (e >= ET) return;
    int s, d;
    if (e < E0) { s = src[e]; d = dst[e]; }
    else        { s = e - E0; d = s; }          // self-loops
    float v = asr[s] + adt[d];
    v = v > 0.0f ? v : NEG_SLOPE * v;
    ebuf[e] = v;
    atomicMax(menc + d, enc_f(v));
}

// ---------------- edge pass 2: w = exp(e - m[dst]); segment sum -----------------
__global__ void k_edge_exp(const int* __restrict__ src, const int* __restrict__ dst,
                           int E0, int ET,
                           const float* __restrict__ ebuf, const unsigned* __restrict__ menc,
                           float* __restrict__ wbuf, float* __restrict__ den) {
    int e = blockIdx.x * blockDim.x + threadIdx.x;
    if (e >= ET) return;
    int d;
    if (e < E0) { d = dst[e]; }
    else        { d = e - E0; }
    float m = dec_f(menc[d]);
    float w = __expf(ebuf[e] - m);
    wbuf[e] = w;
    atomicAdd(den + d, w);
}

// ---------------- edge pass 3: agg[dst] += (w/denom[dst]) * H[src] --------------
// 64 threads per edge, one feature per thread.
__global__ void k_edge_agg(const int* __restrict__ src, const int* __restrict__ dst,
                           int E0, int ET,
                           const float* __restrict__ wbuf, const float* __restrict__ den,
                           const float* __restrict__ H, float* __restrict__ agg) {
    long long t = (long long)blockIdx.x * blockDim.x + threadIdx.x;
    int e = (int)(t >> 6);
    int f = (int)(t & 63);
    if (e >= ET) return;
    int s, d;
    if (e < E0) { s = src[e]; d = dst[e]; }
    else        { s = e - E0; d = s; }
    float alpha = wbuf[e] / (den[d] + 1e-16f);
    atomicAdd(agg + (long long)d * F_OUT + f, alpha * H[(long long)s * F_OUT + f]);
}

// ---------------- bias + optional relu ------------------------------------------
__global__ void k_bias_act(const float* __restrict__ agg, const float* __restrict__ bias,
                           float* __restrict__ out, long long n, int relu) {
    long long t = (long long)blockIdx.x * blockDim.x + threadIdx.x;
    if (t >= n) return;
    float v = agg[t] + bias[(int)(t & 63)];
    if (relu) v = v > 0.0f ? v : 0.0f;
    out[t] = v;
}

// ---------------- host orchestration --------------------------------------------
static void run_layer(const float* Xin, int K, const float* W,
                      const float* avS, const float* avD, const float* bias,
                      const int* src, const int* dst, int N, int E0, int ET,
                      float* H, float* ASRC, float* ADST, float* EBUF, float* WBUF,
                      unsigned* MENC, float* DEN, float* AGG, float* Out, int relu,
                      hipStream_t stream) {
    dim3 gemm_blk(32, 4);
    int mtiles = (N + 15) / 16;
    k_gemm_wmma<<<mtiles, gemm_blk, 0, stream>>>(Xin, W, H, N, K);

    int nb = (N + 255) / 256;
    k_alpha<<<nb, 256, 0, stream>>>(H, avS, avD, ASRC, ADST, N);

    long long nz = (long long)N * F_OUT;
    k_zero<<<(int)((nz + 255) / 256), 256, 0, stream>>>(AGG, nz);
    k_zero<<<nb, 256, 0, stream>>>((float*)MENC, N);
    k_zero<<<nb, 256, 0, stream>>>(DEN, N);

    int eb = (ET + 255) / 256;
    k_edge_max<<<eb, 256, 0, stream>>>(src, dst, E0, ET, ASRC, ADST, EBUF, MENC);
    k_edge_exp<<<eb, 256, 0, stream>>>(src, dst, E0, ET, EBUF, MENC, WBUF, DEN);

    long long et64 = (long long)ET * F_OUT;
    k_edge_agg<<<(int)((et64 + 255) / 256), 256, 0, stream>>>(src, dst, E0, ET,
                                                              WBUF, DEN, H, AGG);
    k_bias_act<<<(int)((nz + 255) / 256), 256, 0, stream>>>(AGG, bias, Out, nz, relu);
}

extern "C" void kernel_launch(void* const* d_in, const int* in_sizes, int n_in,
                              void* d_out, int out_size, void* d_ws, size_t ws_size,
                              hipStream_t stream) {
    const float* x      = (const float*)d_in[0];
    const int*   eidx   = (const int*)d_in[1];
    const float* W1     = (const float*)d_in[2];
    const float* a1_src = (const float*)d_in[3];
    const float* a1_dst = (const float*)d_in[4];
    const float* b1     = (const float*)d_in[5];
    const float* W2     = (const float*)d_in[6];
    const float* a2_src = (const float*)d_in[7];
    const float* a2_dst = (const float*)d_in[8];
    const float* b2     = (const float*)d_in[9];

    const int N  = in_sizes[0] / 128;   // 50000
    const int E0 = in_sizes[1] / 2;     // 800000
    const int ET = E0 + N;              // + self-loops

    const int* src = eidx;
    const int* dst = eidx + E0;

    // workspace carve-up
    char* w = (char*)d_ws;
    float* H    = (float*)w; w += (size_t)N * F_OUT * sizeof(float);  // GEMM result (reused per layer)
    float* HACT = (float*)w; w += (size_t)N * F_OUT * sizeof(float);  // layer-1 activated output
    float* AGG  = (float*)w; w += (size_t)N * F_OUT * sizeof(float);  // softmax aggregation
    float* ASRC = (float*)w; w += (size_t)N * sizeof(float);
    float* ADST = (float*)w; w += (size_t)N * sizeof(float);
    float* EBUF = (float*)w; w += (size_t)ET * sizeof(float);
    float* WBUF = (float*)w; w += (size_t)ET * sizeof(float);
    unsigned* MENC = (unsigned*)w; w += (size_t)N * sizeof(unsigned);
    float* DEN  = (float*)w; w += (size_t)N * sizeof(float);

    // layer 1: X[N,128] @ W1 -> GAT -> relu -> HACT
    run_layer(x, 128, W1, a1_src, a1_dst, b1, src, dst, N, E0, ET,
              H, ASRC, ADST, EBUF, WBUF, MENC, DEN, AGG, HACT, /*relu=*/1, stream);

    // layer 2: HACT[N,64] @ W2 -> GAT -> d_out
    run_layer(HACT, 64, W2, a2_src, a2_dst, b2, src, dst, N, E0, ET,
              H, ASRC, ADST, EBUF, WBUF, MENC, DEN, AGG, (float*)d_out, /*relu=*/0, stream);
}